// TransformMMDetFRCNNRoIHeadOutputs_91173565759784
// MI455X (gfx1250) — compile-verified
//
#include <hip/hip_runtime.h>
#include <math.h>

// ---------------------------------------------------------------------------
// TransformMMDetFRCNNRoIHeadOutputs — Faster R-CNN RoI-head post-processing
// B=8 images, N=2000 proposals, C=80 fg classes (81 logits incl. background)
// Pipeline: softmax+mask -> pack sortable keys -> per-image bitonic top-2000
//           -> gather+decode candidates -> class-offset NMS -> top-100 out
// ---------------------------------------------------------------------------

#define B_IMG   8
#define N_PROP  2000
#define NCLS    80
#define NLOG    81
#define K_PRE   2000
#define M_OUT   100
#define NPAD    262144        // next pow2 >= N_PROP*NCLS = 160000
#define NFLAT   (N_PROP * NCLS)
#define SCORE_THR 0.05f
#define IOU_THR   0.5f
#define MAX_RATIO 4.135166556742356f   // |log(16/1000)|

typedef float v2f_t __attribute__((ext_vector_type(2)));
typedef float v8f_t __attribute__((ext_vector_type(8)));

__device__ __forceinline__ unsigned int ord32(float f) {
  unsigned int u = __float_as_uint(f);
  return (u & 0x80000000u) ? ~u : (u | 0x80000000u);
}

// ---------------------------------------------------------------------------
// Kernel 1: softmax over 81 logits per proposal, write masked fg scores.
// One wave32 handles 16 proposals. The 80-class exp row-sum is computed with
// V_WMMA_F32_16X16X4_F32 against an all-ones B matrix:
//   C[m,n] = sum_k A[m,k] * 1  ->  every column holds the row sum.
// A-operand lane layout (ISA 7.12.2, 32-bit A 16x4): lane l<16 supplies
// A[m=l, K=0..1], lane l>=16 supplies A[m=l-16, K=2..3]. So lane l covers
// classes {4t + 2*(l>>4) + {0,1}} over 20 K-chunks = 40 classes/lane.
// ---------------------------------------------------------------------------
__global__ __launch_bounds__(256) void softmax_scores_kernel(
    const float* __restrict__ cls, float* __restrict__ scores) {
  const int wave = threadIdx.x >> 5;
  const int lane = threadIdx.x & 31;
  const int m    = lane & 15;
  const int hi   = lane >> 4;
  const int row  = (blockIdx.x * 8 + wave) * 16 + m;   // 0..15999
  const float* lg = cls + (size_t)row * NLOG;

  float e0[20], e1[20];
  float bg = lg[80];
  float mymax = bg;
#pragma unroll
  for (int t = 0; t < 20; ++t) {
    const int c0 = 4 * t + 2 * hi;
    e0[t] = lg[c0];
    e1[t] = lg[c0 + 1];
    mymax = fmaxf(mymax, fmaxf(e0[t], e1[t]));
  }
  const float rowmax = fmaxf(mymax, __shfl_xor(mymax, 16, 32));
#pragma unroll
  for (int t = 0; t < 20; ++t) {
    e0[t] = __expf(e0[t] - rowmax);
    e1[t] = __expf(e1[t] - rowmax);
  }

  float rs;  // sum of 80 fg exps for this lane's row
#if __has_builtin(__builtin_amdgcn_wmma_f32_16x16x4_f32)
  {
    v8f_t acc = {};
    v2f_t ones; ones.x = 1.0f; ones.y = 1.0f;
#pragma unroll
    for (int t = 0; t < 20; ++t) {
      v2f_t a; a.x = e0[t]; a.y = e1[t];
      acc = __builtin_amdgcn_wmma_f32_16x16x4_f32(
          false, a, false, ones, (short)0, acc, false, false);
    }
    // D layout: VGPR r, lanes 0-15 -> C[M=r,N=lane] = rowsum(r);
    //           lanes 16-31 -> rowsum(8+r). Stage via LDS to each owner lane.
    __shared__ float rowsum[8][16];
    if (lane == 0) {
#pragma unroll
      for (int r = 0; r < 8; ++r) rowsum[wave][r] = acc[r];
    } else if (lane == 16) {
#pragma unroll
      for (int r = 0; r < 8; ++r) rowsum[wave][8 + r] = acc[r];
    }
    __syncthreads();
    rs = rowsum[wave][m];
  }
#else
  {
    float part = 0.0f;
#pragma unroll
    for (int t = 0; t < 20; ++t) part += e0[t] + e1[t];
    rs = part + __shfl_xor(part, 16, 32);
  }
#endif

  const float inv = 1.0f / (rs + __expf(bg - rowmax));
  float* so = scores + (size_t)row * NCLS;   // flatten order n*C + c
#pragma unroll
  for (int t = 0; t < 20; ++t) {
    const int c0 = 4 * t + 2 * hi;
    const float s0 = e0[t] * inv;
    const float s1 = e1[t] * inv;
    so[c0]     = (s0 > SCORE_THR) ? s0 : -1.0f;
    so[c0 + 1] = (s1 > SCORE_THR) ? s1 : -1.0f;
  }
}

// ---------------------------------------------------------------------------
// Kernel 2: pack 64-bit sortable keys: (ordered(score) << 32) | ~index.
// Descending u64 sort == top_k with lower-index-first stable tie-break.
// Pad [160000, 262144) with key 0 (sorts last).
// ---------------------------------------------------------------------------
__global__ __launch_bounds__(256) void pack_keys_kernel(
    const float* __restrict__ scores, unsigned long long* __restrict__ keys) {
  const int t = blockIdx.x * 256 + threadIdx.x;
  const int b = t >> 18;
  const int i = t & (NPAD - 1);
  unsigned long long key = 0ull;
  if (i < NFLAT) {
    const float s = scores[(size_t)b * NFLAT + i];
    key = ((unsigned long long)ord32(s) << 32) | (unsigned int)(~(unsigned int)i);
  }
  keys[(size_t)b * NPAD + i] = key;
}

// ---------------------------------------------------------------------------
// Kernel 3: per-image bitonic sort (descending) over 262144 keys in global
// memory, one 1024-thread workgroup per image. Deterministic full sort.
// ---------------------------------------------------------------------------
__global__ __launch_bounds__(1024) void bitonic_sort_kernel(
    unsigned long long* __restrict__ keys) {
  const int n = NPAD;
  unsigned long long* a = keys + (size_t)blockIdx.x * n;
  for (int k = 2; k <= n; k <<= 1) {
    for (int j = k >> 1; j > 0; j >>= 1) {
      for (int i = threadIdx.x; i < n; i += 1024) {
        const int l = i ^ j;
        if (l > i) {
          const unsigned long long x = a[i];
          const unsigned long long y = a[l];
          const bool up = (i & k) == 0;           // descending overall
          if (up ? (x < y) : (x > y)) { a[i] = y; a[l] = x; }
        }
      }
      __threadfence();   // make writes visible across waves/CUs in the WGP
      __syncthreads();
    }
  }
}

// ---------------------------------------------------------------------------
// Kernel 4: gather top-2000 candidates, decode their boxes (DeltaXYWH coder),
// and reduce max coordinate (batched-NMS class-offset base).
// ---------------------------------------------------------------------------
__global__ __launch_bounds__(512) void gather_decode_kernel(
    const unsigned long long* __restrict__ keys,
    const float* __restrict__ scores,
    const float* __restrict__ reg, const float* __restrict__ props,
    const int* __restrict__ ph, const int* __restrict__ pw,
    float* __restrict__ cand_score, float* __restrict__ cand_box,
    int* __restrict__ cand_label, float* __restrict__ maxc) {
  const int b = blockIdx.x;
  const float W = (float)(*pw);
  const float H = (float)(*ph);
  float lmax = -3.4e38f;
  for (int k = threadIdx.x; k < K_PRE; k += 512) {
    const unsigned long long key = keys[(size_t)b * NPAD + k];
    const unsigned int idx = ~(unsigned int)key;          // flat n*80 + c
    const int n = (int)(idx / NCLS);
    const int c = (int)(idx - (unsigned int)n * NCLS);
    const float s = scores[(size_t)b * NFLAT + idx];
    const float* roi = props + ((size_t)b * N_PROP + n) * 5;
    const float r0 = roi[0], r1 = roi[1], r2 = roi[2], r3 = roi[3];
    const float px = (r0 + r2) * 0.5f, py = (r1 + r3) * 0.5f;
    const float pwv = r2 - r0, phv = r3 - r1;
    const float* d = reg + ((size_t)b * N_PROP + n) * (NCLS * 4) + c * 4;
    const float dx = d[0] * 0.1f, dy = d[1] * 0.1f;
    const float dw = fminf(fmaxf(d[2] * 0.2f, -MAX_RATIO), MAX_RATIO);
    const float dh = fminf(fmaxf(d[3] * 0.2f, -MAX_RATIO), MAX_RATIO);
    const float gx = px + pwv * dx, gy = py + phv * dy;
    const float gw = pwv * __expf(dw), gh = phv * __expf(dh);
    const float x1 = fminf(fmaxf(gx - gw * 0.5f, 0.0f), W);
    const float y1 = fminf(fmaxf(gy - gh * 0.5f, 0.0f), H);
    const float x2 = fminf(fmaxf(gx + gw * 0.5f, 0.0f), W);
    const float y2 = fminf(fmaxf(gy + gh * 0.5f, 0.0f), H);
    float* cb = cand_box + ((size_t)b * K_PRE + k) * 4;
    cb[0] = x1; cb[1] = y1; cb[2] = x2; cb[3] = y2;
    cand_score[(size_t)b * K_PRE + k] = s;
    cand_label[(size_t)b * K_PRE + k] = c;
    lmax = fmaxf(lmax, fmaxf(fmaxf(x1, y1), fmaxf(x2, y2)));
  }
  __shared__ float smax[512];
  smax[threadIdx.x] = lmax;
  __syncthreads();
  for (int off = 256; off > 0; off >>= 1) {
    if (threadIdx.x < off)
      smax[threadIdx.x] = fmaxf(smax[threadIdx.x], smax[threadIdx.x + off]);
    __syncthreads();
  }
  if (threadIdx.x == 0) maxc[b] = smax[0];
}

// ---------------------------------------------------------------------------
// Kernel 5: class-offset greedy NMS in LDS + final top-100.
// LDS: shifted boxes (4x8000B) + area (8000B) + keep (8000B) + sort keys
// (16384B) = 64384 B.
// ---------------------------------------------------------------------------
__global__ __launch_bounds__(1024) void nms_finalize_kernel(
    const float* __restrict__ cand_score, const float* __restrict__ cand_box,
    const int* __restrict__ cand_label, const float* __restrict__ maxc,
    float* __restrict__ out) {
  const int b = blockIdx.x;
  __shared__ float sx1[K_PRE], sy1[K_PRE], sx2[K_PRE], sy2[K_PRE];
  __shared__ float sarea[K_PRE];
  __shared__ int   skeep[K_PRE];
  __shared__ unsigned long long skey[2048];

  const float offmul = maxc[b] + 1.0f;
  for (int k = threadIdx.x; k < K_PRE; k += 1024) {
    const float* bb = cand_box + ((size_t)b * K_PRE + k) * 4;
    const float o = (float)cand_label[(size_t)b * K_PRE + k] * offmul;
    const float x1 = bb[0] + o, y1 = bb[1] + o, x2 = bb[2] + o, y2 = bb[3] + o;
    sx1[k] = x1; sy1[k] = y1; sx2[k] = x2; sy2[k] = y2;
    sarea[k] = (x2 - x1) * (y2 - y1);
    skeep[k] = (cand_score[(size_t)b * K_PRE + k] > SCORE_THR) ? 1 : 0;
  }
  __syncthreads();

  // Greedy NMS: sequential over i (candidates are score-sorted), parallel j.
  for (int i = 0; i < K_PRE - 1; ++i) {
    if (skeep[i]) {
      const float xi1 = sx1[i], yi1 = sy1[i], xi2 = sx2[i], yi2 = sy2[i];
      const float ai = sarea[i];
      for (int j = i + 1 + threadIdx.x; j < K_PRE; j += 1024) {
        if (skeep[j]) {
          const float ix1 = fmaxf(xi1, sx1[j]);
          const float iy1 = fmaxf(yi1, sy1[j]);
          const float ix2 = fminf(xi2, sx2[j]);
          const float iy2 = fminf(yi2, sy2[j]);
          const float iw = fmaxf(ix2 - ix1, 0.0f);
          const float ih = fmaxf(iy2 - iy1, 0.0f);
          const float inter = iw * ih;
          const float iou = inter / (ai + sarea[j] - inter + 1e-6f);
          if (iou > IOU_THR) skeep[j] = 0;
        }
      }
    }
    __syncthreads();
  }

  // Final top-100 via 2048-element LDS bitonic (descending, stable ties).
  for (int k = threadIdx.x; k < 2048; k += 1024) {
    unsigned long long key = 0ull;
    if (k < K_PRE) {
      const float f = skeep[k] ? cand_score[(size_t)b * K_PRE + k] : -1.0f;
      key = ((unsigned long long)ord32(f) << 32) | (unsigned int)(~(unsigned int)k);
    }
    skey[k] = key;
  }
  __syncthreads();
  for (int k = 2; k <= 2048; k <<= 1) {
    for (int j = k >> 1; j > 0; j >>= 1) {
      for (int i = threadIdx.x; i < 2048; i += 1024) {
        const int l = i ^ j;
        if (l > i) {
          const unsigned long long x = skey[i];
          const unsigned long long y = skey[l];
          const bool up = (i & k) == 0;
          if (up ? (x < y) : (x > y)) { skey[i] = y; skey[l] = x; }
        }
      }
      __syncthreads();
    }
  }

  if (threadIdx.x < M_OUT) {
    const int t = threadIdx.x;
    const unsigned long long key = skey[t];
    const int ki = (int)(~(unsigned int)key);
    const float s = skeep[ki] ? cand_score[(size_t)b * K_PRE + ki] : -1.0f;
    const float* bb = cand_box + ((size_t)b * K_PRE + ki) * 4;
    float* det = out + ((size_t)b * M_OUT + t) * 5;
    det[0] = bb[0]; det[1] = bb[1]; det[2] = bb[2]; det[3] = bb[3]; det[4] = s;
    out[(size_t)B_IMG * M_OUT * 5 + (size_t)b * M_OUT + t] =
        (s > SCORE_THR) ? (float)cand_label[(size_t)b * K_PRE + ki] : -1.0f;
  }
}

// ---------------------------------------------------------------------------
extern "C" void kernel_launch(void* const* d_in, const int* in_sizes, int n_in,
                              void* d_out, int out_size, void* d_ws, size_t ws_size,
                              hipStream_t stream) {
  const float* cls   = (const float*)d_in[0];   // [8,2000,81]
  const float* reg   = (const float*)d_in[1];   // [8,2000,320]
  const float* props = (const float*)d_in[2];   // [8,2000,5]
  const int*   ph    = (const int*)d_in[3];     // img_h
  const int*   pw    = (const int*)d_in[4];     // img_w
  float* out = (float*)d_out;                   // 4000 det floats + 800 labels

  char* ws = (char*)d_ws;
  size_t off = 0;
  float* ws_scores = (float*)(ws + off);                 off += (size_t)B_IMG * NFLAT * 4;      // 5.12 MB
  unsigned long long* ws_keys = (unsigned long long*)(ws + off);
                                                         off += (size_t)B_IMG * NPAD * 8;       // 16.78 MB
  float* cand_score = (float*)(ws + off);                off += (size_t)B_IMG * K_PRE * 4;
  float* cand_box   = (float*)(ws + off);                off += (size_t)B_IMG * K_PRE * 4 * 4;
  int*   cand_label = (int*)(ws + off);                  off += (size_t)B_IMG * K_PRE * 4;
  float* maxcoord   = (float*)(ws + off);                off += (size_t)B_IMG * 4;
  (void)ws_size; (void)in_sizes; (void)n_in; (void)out_size;  // ~22.3 MB total ws use

  // 1) softmax + score mask (WMMA row-sum): 16000 rows / 16 per wave / 8 waves per block
  softmax_scores_kernel<<<125, 256, 0, stream>>>(cls, ws_scores);
  // 2) pack sortable keys, pad to 2^18 per image
  pack_keys_kernel<<<(B_IMG * NPAD) / 256, 256, 0, stream>>>(ws_scores, ws_keys);
  // 3) per-image descending bitonic sort (top-2000 = first 2000 keys)
  bitonic_sort_kernel<<<B_IMG, 1024, 0, stream>>>(ws_keys);
  // 4) gather + decode candidates, max-coordinate reduction
  gather_decode_kernel<<<B_IMG, 512, 0, stream>>>(ws_keys, ws_scores, reg, props,
                                                  ph, pw, cand_score, cand_box,
                                                  cand_label, maxcoord);
  // 5) class-offset NMS + top-100 output
  nms_finalize_kernel<<<B_IMG, 1024, 0, stream>>>(cand_score, cand_box, cand_label,
                                                  maxcoord, out);
}